// AttentiveMinkUNetDiff_78958678769887
// MI455X (gfx1250) — compile-verified
//
#include <hip/hip_runtime.h>
#include <hip/hip_bf16.h>
#include <math.h>

#define N_NODES 16384
#define M_COND  32768
#define KNN     8
#define C_CONDF 256
#define C_STG   256
#define EMBED   96
#define EMBED_H 48

typedef float v2f __attribute__((ext_vector_type(2)));
typedef float v8f __attribute__((ext_vector_type(8)));

// ---------------------------------------------------------------- max reduce
__global__ void init_max_kernel(int* maxp) {
    if (threadIdx.x == 0) *maxp = 0;
}

__global__ void max_reduce_kernel(const int* __restrict__ coords, int n, int* __restrict__ maxp) {
    __shared__ int sm[256];
    int tid = threadIdx.x;
    int stride = gridDim.x * blockDim.x;
    int v = 0;
    for (int i = blockIdx.x * blockDim.x + tid; i < n; i += stride)
        v = max(v, coords[i]);
    sm[tid] = v;
    __syncthreads();
    for (int s = blockDim.x >> 1; s > 0; s >>= 1) {
        if (tid < s) sm[tid] = max(sm[tid], sm[tid + s]);
        __syncthreads();
    }
    if (tid == 0) atomicMax(maxp, sm[0]);
}

// ---------------------------------------------------------------- coord transform
__global__ void transform_kernel(const int* __restrict__ coords, int n,
                                 float stride_, float voxel,
                                 const int* __restrict__ maxp,
                                 float4* __restrict__ out) {
    int i = blockIdx.x * blockDim.x + threadIdx.x;
    if (i >= n) return;
    float maxc = (float)(*maxp);
    int4 c = ((const int4*)coords)[i];
    float4 o;
    o.x = (float)c.x * (maxc * 2.0f);
    o.y = ((float)c.y + stride_ * 0.5f) * voxel;
    o.z = ((float)c.z + stride_ * 0.5f) * voxel;
    o.w = ((float)c.w + stride_ * 0.5f) * voxel;
    out[i] = o;
}

// ---------------------------------------------------------------- KNN (top-8)
// Cond-point tiles are staged into LDS with the CDNA5 async-to-LDS path
// (GLOBAL_LOAD_ASYNC_TO_LDS_B128, ASYNCcnt-tracked, no VGPR round-trip).
#define KNN_TILE 256
__global__ __launch_bounds__(128)
void knn_kernel(const float4* __restrict__ full_c, const float4* __restrict__ part_c,
                int* __restrict__ out_idx, float* __restrict__ out_w) {
    __shared__ float4 tile[KNN_TILE];
    int n = blockIdx.x * blockDim.x + threadIdx.x;   // one node per thread
    float4 me = full_c[n];

    float bd[KNN];
    int   bi[KNN];
#pragma unroll
    for (int j = 0; j < KNN; ++j) { bd[j] = 3.4e38f; bi[j] = 0; }

    for (int base = 0; base < M_COND; base += KNN_TILE) {
        __syncthreads();   // previous tile fully consumed
#if defined(__gfx1250__)
        for (int t = threadIdx.x; t < KNN_TILE; t += blockDim.x) {
            // flat LDS address: low 32 bits are the LDS byte offset (ISA 10.2)
            uint32_t lds_off = (uint32_t)(uintptr_t)(&tile[t]);
            uint64_t gaddr   = (uint64_t)(uintptr_t)(part_c + base + t);
            asm volatile("global_load_async_to_lds_b128 %0, %1, off"
                         :: "v"(lds_off), "v"(gaddr) : "memory");
        }
        asm volatile("s_wait_asynccnt 0x0" ::: "memory");
#else
        for (int t = threadIdx.x; t < KNN_TILE; t += blockDim.x)
            tile[t] = part_c[base + t];
#endif
        __syncthreads();
#pragma unroll 4
        for (int t = 0; t < KNN_TILE; ++t) {
            float4 p = tile[t];
            float db = p.x - me.x, dx = p.y - me.y, dy = p.z - me.z, dz = p.w - me.w;
            float d2 = fmaf(db, db, fmaf(dx, dx, fmaf(dy, dy, dz * dz)));
            if (d2 < bd[KNN - 1]) {   // strict < keeps earlier index on ties (top_k stability)
                float dc = d2; int ic = base + t;
#pragma unroll
                for (int j = 0; j < KNN; ++j) {
                    bool sw = dc < bd[j];
                    float td = bd[j]; int ti = bi[j];
                    bd[j] = sw ? dc : td;  bi[j] = sw ? ic : ti;
                    dc    = sw ? td : dc;  ic    = sw ? ti : ic;
                }
            }
        }
    }

    // inverse-distance weights: dists use xyz only (reference drops batch dim)
    float wk[KNN];
    float wsum = 0.0f;
#pragma unroll
    for (int j = 0; j < KNN; ++j) {
        float4 p = part_c[bi[j]];
        float dx = p.y - me.y, dy = p.z - me.z, dz = p.w - me.w;
        float d = sqrtf(fmaf(dx, dx, fmaf(dy, dy, dz * dz)));
        d = fmaxf(d, 1e-6f);
        wk[j] = 1.0f / d;
        wsum += wk[j];
    }
    float inv = 1.0f / wsum;
#pragma unroll
    for (int j = 0; j < KNN; ++j) {
        out_idx[n * KNN + j] = bi[j];
        out_w[n * KNN + j]   = wk[j] * inv;
    }
}

// ---------------------------------------------------------------- weighted aggregation
__global__ __launch_bounds__(256)
void aggregate_kernel(const float* __restrict__ feats, const int* __restrict__ idx,
                      const float* __restrict__ w, float* __restrict__ agg) {
    int n = blockIdx.x;
    int c = threadIdx.x;
    float acc = 0.0f;
#pragma unroll
    for (int k = 0; k < KNN; ++k) {
        int j   = idx[n * KNN + k];
        float v = w[n * KNN + k];
        acc = fmaf(v, feats[j * C_CONDF + c], acc);
    }
    agg[n * C_CONDF + c] = acc;
}

// ---------------------------------------------------------------- timestep MLP (tiny, 1 block)
__global__ __launch_bounds__(256)
void tfeat_kernel(const float* __restrict__ t,
                  const float* __restrict__ Wt1, const float* __restrict__ bt1,
                  const float* __restrict__ Wt2, const float* __restrict__ bt2,
                  float* __restrict__ tf) {
    __shared__ float emb[EMBED];
    __shared__ float hid[EMBED];
    int tid = threadIdx.x;
    float tv = t[0];
    if (tid < EMBED) {
        int i = (tid < EMBED_H) ? tid : (tid - EMBED_H);
        float f = expf((float)i * (-logf(10000.0f) / (float)(EMBED_H - 1)));
        float e = tv * f;
        emb[tid] = (tid < EMBED_H) ? sinf(e) : cosf(e);
    }
    __syncthreads();
    if (tid < EMBED) {
        float acc = bt1[tid];
        for (int j = 0; j < EMBED; ++j) acc = fmaf(emb[j], Wt1[tid * EMBED + j], acc);
        hid[tid] = acc > 0.0f ? acc : 0.1f * acc;
    }
    __syncthreads();
    if (tid < C_STG) {
        float acc = bt2[tid];
        for (int j = 0; j < EMBED; ++j) acc = fmaf(hid[j], Wt2[tid * EMBED + j], acc);
        tf[tid] = acc;
    }
}

// ---------------------------------------------------------------- fp32 WMMA GEMM
// Y[nrows x 256] = act( X[nrows x 256] @ W^T + bias (+ addvec) )
// One wave computes a 32(M) x 64(N) strip: two M-tiles share the 4 B-fragments,
// halving B traffic per WMMA vs a 16x64 strip. 8 accumulators = 64 VGPRs.
// V_WMMA_F32_16X16X4_F32 fragment layout (ISA 7.12.2, 32-bit):
//   A: lane(0..15)=row m, vgpr0/1 = K+0/K+1 ; lane(16..31)=row m, K+2/K+3
//   B: lane(0..15)=col n, vgpr0/1 = K+0/K+1 ; lane(16..31)=col n, K+2/K+3
//   C: vgpr r holds rows r (lanes 0-15) and r+8 (lanes 16-31), col = lane%16
__global__ __launch_bounds__(256)
void gemm_wmma_kernel(const float* __restrict__ X, const float* __restrict__ W,
                      const float* __restrict__ bias, const float* __restrict__ addvec,
                      float* __restrict__ Y, int nrows, int leaky) {
    const int lane = threadIdx.x & 31;
    const int wave = threadIdx.x >> 5;
    const int tile = blockIdx.x * 8 + wave;     // 8 waves per block
    const int tiles_n = C_STG / 64;             // 4 col-strips of 64
    const int m0 = (tile / tiles_n) * 32;
    const int d0 = (tile % tiles_n) * 64;
    if (m0 >= nrows) return;

    const int lrow  = lane & 15;
    const int khalf = (lane >> 4) << 1;         // 0 for lanes 0-15, 2 for 16-31
    const int mhi   = (lane >> 4) * 8;

    const float* xrow0 = X + (m0 + lrow) * C_STG;
    const float* xrow1 = X + (m0 + 16 + lrow) * C_STG;
    const float* wr[4];
#pragma unroll
    for (int s = 0; s < 4; ++s) wr[s] = W + (d0 + s * 16 + lrow) * C_STG;

    __builtin_prefetch(xrow0, 0, 1);
    __builtin_prefetch(xrow1, 0, 1);
#pragma unroll
    for (int s = 0; s < 4; ++s) __builtin_prefetch(wr[s], 0, 1);

    v8f acc[2][4];
#pragma unroll
    for (int h = 0; h < 2; ++h)
#pragma unroll
        for (int s = 0; s < 4; ++s) acc[h][s] = (v8f){};

#pragma unroll 4
    for (int k = 0; k < C_STG; k += 4) {
        const int kb = k + khalf;
        v2f a0, a1;
        a0.x = xrow0[kb];  a0.y = xrow0[kb + 1];
        a1.x = xrow1[kb];  a1.y = xrow1[kb + 1];
#pragma unroll
        for (int s = 0; s < 4; ++s) {
            v2f b;
            b.x = wr[s][kb];
            b.y = wr[s][kb + 1];
            acc[0][s] = __builtin_amdgcn_wmma_f32_16x16x4_f32(
                false, a0, false, b, (short)0, acc[0][s], false, false);
            acc[1][s] = __builtin_amdgcn_wmma_f32_16x16x4_f32(
                false, a1, false, b, (short)0, acc[1][s], false, false);
        }
    }

#pragma unroll
    for (int s = 0; s < 4; ++s) {
        int col = d0 + s * 16 + lrow;
        float bv = bias[col] + (addvec ? addvec[col] : 0.0f);
#pragma unroll
        for (int h = 0; h < 2; ++h) {
#pragma unroll
            for (int r = 0; r < 8; ++r) {
                float v = acc[h][s][r] + bv;
                if (leaky) v = v > 0.0f ? v : 0.1f * v;
                Y[(m0 + 16 * h + r + mhi) * C_STG + col] = v;
            }
        }
    }
}

// ---------------------------------------------------------------- host launch
extern "C" void kernel_launch(void* const* d_in, const int* in_sizes, int n_in,
                              void* d_out, int out_size, void* d_ws, size_t ws_size,
                              hipStream_t stream) {
    const int*   node_coords = (const int*)d_in[0];
    const int*   cond_coords = (const int*)d_in[1];
    const float* cond_feats  = (const float*)d_in[2];
    const float* t_in        = (const float*)d_in[3];
    const float* W_proj = (const float*)d_in[4];
    const float* b_proj = (const float*)d_in[5];
    const float* W_l1   = (const float*)d_in[6];
    const float* b_l1   = (const float*)d_in[7];
    const float* W_l2   = (const float*)d_in[8];
    const float* b_l2   = (const float*)d_in[9];
    const float* W_t1   = (const float*)d_in[10];
    const float* b_t1   = (const float*)d_in[11];
    const float* W_t2   = (const float*)d_in[12];
    const float* b_t2   = (const float*)d_in[13];

    // workspace layout (bytes)
    char* ws = (char*)d_ws;
    constexpr size_t OFF_MAX   = 0;
    constexpr size_t OFF_TF    = 1024;
    constexpr size_t OFF_FULLC = 8192;
    constexpr size_t OFF_PARTC = OFF_FULLC + (size_t)N_NODES * 16;
    constexpr size_t OFF_IDX   = OFF_PARTC + (size_t)M_COND * 16;
    constexpr size_t OFF_W     = OFF_IDX + (size_t)N_NODES * KNN * 4;
    constexpr size_t OFF_AGG   = OFF_W + (size_t)N_NODES * KNN * 4;

    int*    maxp   = (int*)   (ws + OFF_MAX);
    float*  tf     = (float*) (ws + OFF_TF);
    float4* full_c = (float4*)(ws + OFF_FULLC);
    float4* part_c = (float4*)(ws + OFF_PARTC);
    int*    idx    = (int*)   (ws + OFF_IDX);
    float*  wgt    = (float*) (ws + OFF_W);
    float*  agg    = (float*) (ws + OFF_AGG);
    float*  out    = (float*)d_out;

    hipLaunchKernelGGL(init_max_kernel, dim3(1), dim3(1), 0, stream, maxp);
    hipLaunchKernelGGL(max_reduce_kernel, dim3(64), dim3(256), 0, stream,
                       node_coords, N_NODES * 4, maxp);

    hipLaunchKernelGGL(transform_kernel, dim3((N_NODES + 255) / 256), dim3(256), 0, stream,
                       node_coords, N_NODES, 16.0f, 0.05f, maxp, full_c);
    hipLaunchKernelGGL(transform_kernel, dim3((M_COND + 255) / 256), dim3(256), 0, stream,
                       cond_coords, M_COND, 1.0f, 0.01f, maxp, part_c);

    hipLaunchKernelGGL(knn_kernel, dim3(N_NODES / 128), dim3(128), 0, stream,
                       full_c, part_c, idx, wgt);

    hipLaunchKernelGGL(aggregate_kernel, dim3(N_NODES), dim3(256), 0, stream,
                       cond_feats, idx, wgt, agg);

    hipLaunchKernelGGL(tfeat_kernel, dim3(1), dim3(256), 0, stream,
                       t_in, W_t1, b_t1, W_t2, b_t2, tf);

    // GEMM chain: agg -> d_out(matches) -> agg(hidden) -> d_out(final)
    const int gemm_blocks = (N_NODES / 32) * (C_STG / 64) / 8;   // 256
    hipLaunchKernelGGL(gemm_wmma_kernel, dim3(gemm_blocks), dim3(256), 0, stream,
                       agg, W_proj, b_proj, (const float*)nullptr, out, N_NODES, 0);
    hipLaunchKernelGGL(gemm_wmma_kernel, dim3(gemm_blocks), dim3(256), 0, stream,
                       out, W_l1, b_l1, (const float*)nullptr, agg, N_NODES, 1);
    hipLaunchKernelGGL(gemm_wmma_kernel, dim3(gemm_blocks), dim3(256), 0, stream,
                       agg, W_l2, b_l2, tf, out, N_NODES, 0);
}